// ExLoss_58944131170501
// MI455X (gfx1250) — compile-verified
//
#include <hip/hip_runtime.h>
#include <hip/hip_bf16.h>
#include <math.h>

// Problem shape (fixed by the reference setup)
#define N_ROWS 128
#define CDIM   2048
#define KPOS   4
#define KNEG   508
#define KTOT   (KPOS + KNEG)   // 512

typedef __attribute__((ext_vector_type(2))) float v2f;
typedef __attribute__((ext_vector_type(8))) float v8f;

// ---------------------------------------------------------------------------
// Kernel 1: gathered matvec via V_WMMA_F32_16X16X4_F32.
// One block per output row n.  8 waves/block; each wave computes 4 tiles of
// 16 logits.  A-matrix = 16 gathered M rows x 4 c-values (per-lane float2
// streaming loads); B-matrix = inputs[n, c..c+3] broadcast across all 16
// columns (read from LDS).  All 16 D columns are identical == the dot
// products; column N=0 is extracted from lanes 0 and 16.
// ---------------------------------------------------------------------------
__global__ __launch_bounds__(256) void gather_gemm_wmma(
    const float* __restrict__ inputs,   // [N_ROWS, CDIM]
    const int*   __restrict__ posIdx,   // [N_ROWS, KPOS]
    const int*   __restrict__ negIdx,   // [N_ROWS, KNEG]
    const float* __restrict__ M,        // [NC, CDIM]
    float*       __restrict__ logits)   // [N_ROWS, KTOT]
{
    __shared__ float sx[CDIM];          // 8 KB: inputs[n, :]
    const int n = blockIdx.x;

    // Stage inputs[n] into LDS (coalesced float4 loads)
    const float4* xin4 = (const float4*)(inputs + (size_t)n * CDIM);
    float4* sx4 = (float4*)sx;
    for (int i = threadIdx.x; i < CDIM / 4; i += blockDim.x) sx4[i] = xin4[i];
    __syncthreads();

    const int lane = threadIdx.x & 31;
    const int wave = threadIdx.x >> 5;
    const int row  = lane & 15;          // A-matrix M index within the tile
    const int half = lane >> 4;          // 0: K=0,1   1: K=2,3
    const int coff = 2 * half;

    // 32 tiles of 16 logits, 8 waves -> 4 tiles per wave
    for (int tile = wave; tile < KTOT / 16; tile += 8) {
        const int k0 = tile * 16;
        const int k  = k0 + row;
        // concat(positive_index, negative_index) gather
        const int gidx = (k < KPOS) ? posIdx[n * KPOS + k]
                                    : negIdx[n * KNEG + (k - KPOS)];
        const float* __restrict__ arow = M + (size_t)gidx * CDIM + coff;
        const float* __restrict__ brow = sx + coff;

        v8f acc = {};
        #pragma unroll 8
        for (int c = 0; c < CDIM; c += 4) {
            v2f a = *(const v2f*)(arow + c);   // A[row, coff..coff+1]
            v2f b = *(const v2f*)(brow + c);   // B broadcast of inputs[n, c+coff..]
            // (neg_a, A, neg_b, B, c_mod, C, reuse_a, reuse_b)
            acc = __builtin_amdgcn_wmma_f32_16x16x4_f32(
                false, a, false, b, (short)0, acc, false, false);
        }

        // D column N=0: lane 0 holds M=0..7 in acc[0..7]; lane 16 holds M=8..15
        float* out = logits + (size_t)n * KTOT + k0;
        if (lane == 0) {
            #pragma unroll
            for (int r = 0; r < 8; ++r) out[r] = acc[r];
        }
        if (lane == 16) {
            #pragma unroll
            for (int r = 0; r < 8; ++r) out[8 + r] = acc[r];
        }
    }
}

// ---------------------------------------------------------------------------
// Kernel 2: per-row log-sum-exp + partial loss.
// One block (512 threads) per row; deterministic LDS tree reductions.
// rowpart[n] = sum_j cof[j] * (logits[n,j] - lse_n)
// ---------------------------------------------------------------------------
__global__ __launch_bounds__(512) void row_lse(
    const float* __restrict__ logits,   // [N_ROWS, KTOT]
    const float* __restrict__ cof,      // [KPOS]
    float*       __restrict__ rowpart)  // [N_ROWS]
{
    __shared__ float red[KTOT];
    const int n = blockIdx.x;
    const int t = threadIdx.x;
    const float* rowp = logits + (size_t)n * KTOT;

    const float x = rowp[t];
    red[t] = x;
    __syncthreads();
    for (int s = KTOT / 2; s > 0; s >>= 1) {
        if (t < s) red[t] = fmaxf(red[t], red[t + s]);
        __syncthreads();
    }
    const float m = red[0];
    __syncthreads();

    red[t] = __expf(x - m);
    __syncthreads();
    for (int s = KTOT / 2; s > 0; s >>= 1) {
        if (t < s) red[t] += red[t + s];
        __syncthreads();
    }

    if (t == 0) {
        const float lse = __logf(red[0]) + m;
        float p = 0.0f;
        #pragma unroll
        for (int j = 0; j < KPOS; ++j) p += cof[j] * (rowp[j] - lse);
        rowpart[n] = p;
    }
}

// ---------------------------------------------------------------------------
// Kernel 3: fixed-order reduction of the 128 row partials -> scalar loss.
// ---------------------------------------------------------------------------
__global__ __launch_bounds__(128) void finalize_loss(
    const float* __restrict__ rowpart, float* __restrict__ loss)
{
    __shared__ float red[N_ROWS];
    const int t = threadIdx.x;
    red[t] = rowpart[t];
    __syncthreads();
    for (int s = N_ROWS / 2; s > 0; s >>= 1) {
        if (t < s) red[t] += red[t + s];
        __syncthreads();
    }
    if (t == 0) loss[0] = -red[0] / (float)N_ROWS;
}

// ---------------------------------------------------------------------------
extern "C" void kernel_launch(void* const* d_in, const int* in_sizes, int n_in,
                              void* d_out, int out_size, void* d_ws, size_t ws_size,
                              hipStream_t stream) {
    (void)in_sizes; (void)n_in; (void)out_size; (void)ws_size;

    const float* inputs = (const float*)d_in[0];   // [128, 2048] f32
    const int*   posIdx = (const int*)  d_in[1];   // [128, 4]
    const int*   negIdx = (const int*)  d_in[2];   // [128, 508]
    const float* cof    = (const float*)d_in[3];   // [4] f32
    const float* M      = (const float*)d_in[4];   // [50000, 2048] f32

    float* out    = (float*)d_out;                 // [0]=loss, [1..65536]=logits
    float* logits = out + 1;
    float* rowpart = (float*)d_ws;                 // 128 floats of scratch

    gather_gemm_wmma<<<N_ROWS, 256, 0, stream>>>(inputs, posIdx, negIdx, M, logits);
    row_lse<<<N_ROWS, KTOT, 0, stream>>>(logits, cof, rowpart);
    finalize_loss<<<1, N_ROWS, 0, stream>>>(rowpart, out);
}